// OrthogonalLatentParameterization_55817394979605
// MI455X (gfx1250) — compile-verified
//
#include <hip/hip_runtime.h>
#include <hip/hip_bf16.h>

// out^T = expm(S) v, S = raw - raw^T  (skew), v = base_noise^T
// Taylor action: acc = v; w = v; for k=1..K: w = S w / k; acc += w
// Each S·w = raw·w - raw^T·w done with V_WMMA_F32_16X16X4_F32 tiles (exact f32).

#define ZDIM 2048
#define NTERMS 8
#define WAVES_PER_BLK 8   // 256 threads = 8 wave32

typedef __attribute__((ext_vector_type(2))) float v2f;
typedef __attribute__((ext_vector_type(8))) float v8f;

__global__ __launch_bounds__(256) void olp_init_kernel(const float* __restrict__ base,
                                                       float* __restrict__ out,
                                                       float* __restrict__ w, int n) {
    int i = blockIdx.x * blockDim.x + threadIdx.x;
    if (i < n) {
        float v = base[i];
        out[i] = v;   // acc starts at v (k=0 term)
        w[i]   = v;   // current Taylor term
    }
}

// One wave computes 16 outputs m0..m0+15:
//   y1[m] = sum_k raw[m,k]   * w[k]   (A-side WMMA: A = raw tile, B = w broadcast)
//   y2[m] = sum_k raw[k,m]   * w[k]   (B-side WMMA: A = w broadcast, B = raw tile)
//   u[m]  = (y1[m] - y2[m]) * inv_k ; w_out[m] = u ; acc[m] += u
__global__ __launch_bounds__(256) void olp_taylor_step_kernel(
        const float* __restrict__ raw,
        const float* __restrict__ w_in,
        float* __restrict__ w_out,
        float* __restrict__ acc,
        float inv_k) {
    __shared__ float wlds[ZDIM];
    __shared__ float y1buf[WAVES_PER_BLK][16];
    __shared__ float y2buf[WAVES_PER_BLK][16];

    const int tid = threadIdx.x;
    // Stage the whole vector in LDS (8 KB), coalesced.
    for (int i = tid; i < ZDIM; i += 256) wlds[i] = w_in[i];
    __syncthreads();

    const int wave = tid >> 5;
    const int lane = tid & 31;
    const int lo   = lane & 15;       // M (A-side) / N (B-side) index
    const int h    = lane >> 4;       // half selector: K pair {2h, 2h+1}
    const int m0   = (blockIdx.x * WAVES_PER_BLK + wave) * 16;

    const float* __restrict__ arow = raw + (size_t)(m0 + lo) * ZDIM;

    v8f c1 = {};  // accumulates y1 (replicated over N)
    v8f c2 = {};  // accumulates y2 (replicated over M)

    // ISA 7.12.2 layout for 32-bit 16x4 A / 4x16 B:
    //   vgpr0 holds K = 2h, vgpr1 holds K = 2h+1  (h = lane>=16)
    #pragma unroll 4
    for (int k0 = 0; k0 < ZDIM; k0 += 4) {
        const int kk = k0 + 2 * h;

        v2f wv;                       // w broadcast operand (valid as A or B)
        wv.x = wlds[kk];
        wv.y = wlds[kk + 1];

        v2f a;                        // A tile: raw[m0+lo, kk..kk+1]
        a.x = arow[kk];
        a.y = arow[kk + 1];

        v2f b;                        // B tile: raw[kk..kk+1, m0+lo] (coalesced)
        b.x = raw[(size_t)kk * ZDIM + m0 + lo];
        b.y = raw[(size_t)(kk + 1) * ZDIM + m0 + lo];

        // (neg_a, A, neg_b, B, c_mod, C, reuse_a, reuse_b)
        c1 = __builtin_amdgcn_wmma_f32_16x16x4_f32(false, a,  false, wv, (short)0, c1, false, false);
        c2 = __builtin_amdgcn_wmma_f32_16x16x4_f32(false, wv, false, b,  (short)0, c2, false, false);
    }

    // D layout (32-bit 16x16): lane<16 -> N=lane, M=vgpr; lane>=16 -> N=lane-16, M=vgpr+8.
    // c1 replicated over N: lane 0 has y1[m0+0..7], lane 16 has y1[m0+8..15].
    if (lane == 0) {
        #pragma unroll
        for (int j = 0; j < 8; ++j) y1buf[wave][j] = c1[j];
    }
    if (lane == 16) {
        #pragma unroll
        for (int j = 0; j < 8; ++j) y1buf[wave][8 + j] = c1[j];
    }
    // c2 replicated over M: lane L (<16) holds y2[m0+L] in every C vgpr.
    if (lane < 16) y2buf[wave][lo] = c2[0];
    __syncthreads();

    if (lane < 16) {
        const int m = m0 + lane;
        const float u = (y1buf[wave][lane] - y2buf[wave][lane]) * inv_k;
        w_out[m] = u;
        acc[m]  += u;   // sole writer of this element in this kernel
    }
}

extern "C" void kernel_launch(void* const* d_in, const int* in_sizes, int n_in,
                              void* d_out, int out_size, void* d_ws, size_t ws_size,
                              hipStream_t stream) {
    const float* raw  = (const float*)d_in[0];   // (2048, 2048) f32
    const float* base = (const float*)d_in[1];   // (1, 2048)   f32
    float* out = (float*)d_out;                  // (1, 2048)   f32  == acc
    float* wA  = (float*)d_ws;                   // ping
    float* wB  = wA + ZDIM;                      // pong   (16 KB total scratch)

    olp_init_kernel<<<(ZDIM + 255) / 256, 256, 0, stream>>>(base, out, wA, ZDIM);

    float* win  = wA;
    float* wout = wB;
    for (int k = 1; k <= NTERMS; ++k) {
        // 16 blocks x 8 waves x 16 rows = 2048 outputs
        olp_taylor_step_kernel<<<ZDIM / (16 * WAVES_PER_BLK), 256, 0, stream>>>(
            raw, win, wout, out, 1.0f / (float)k);
        float* t = win; win = wout; wout = t;
    }
}